// FastSpeech_77524159693213
// MI455X (gfx1250) — compile-verified
//
#include <hip/hip_runtime.h>
#include <math.h>

// ---------------- model dims ----------------
#define Bc   4
#define Sc   512
#define Hc   256
#define Fc   1024
#define NHc  2
#define HCc  128
#define Lc   2048
#define OUTc 80
#define KSc  3

typedef float v2f __attribute__((ext_vector_type(2)));
typedef float v8f __attribute__((ext_vector_type(8)));

#define FLAG_TRANSB 1
#define FLAG_BIAS   2
#define FLAG_ACC    4
#define FLAG_TRANSC 8

// =====================================================================
// Generic fp32 WMMA GEMM: one wave32 computes a 16x32 tile of C
// (two 16x16 WMMA sub-tiles sharing one A fragment).
//   C(z)[m,n] = alpha * ( init + sum_k A(z)[m+rowShiftA, k] * B(z)[k,n] )
// init = 0 | bias[n] | C(z)[m,n]  (flags)
// Batched offsets: ptr += (z/zdiv)*Hi + (z%zdiv)*Lo
// B element (k,n): transB ? B[n*ldb + k*ldbe] : B[k*ldb + n*ldbe]
// Fragment layouts per CDNA5 ISA 7.12.2 (A 16x4 = 2 VGPR/lane; B/C/D row-
// striped across lanes; lanes 16..31 hold K=2,3 / M=8..15).
// K-loop is 2-stage software pipelined. A-row zero-padding: address is
// clamped to a valid row, loads are UNCONDITIONAL, and the loaded value is
// zeroed with a select (v_cndmask) — no divergent branches in the loop.
// =====================================================================
__global__ __launch_bounds__(32)
void wmma_gemm_f32(const float* __restrict__ A, const float* __restrict__ B,
                   const float* __restrict__ bias, float* __restrict__ C,
                   int M, int N, int K,
                   int lda, int ldb, int ldbe, int ldc,
                   long aHi, long aLo, long bHi, long bLo, long cHi, long cLo,
                   int zdiv, int rowShiftA, float alpha, int flags)
{
    const int lane = threadIdx.x;
    const int l15  = lane & 15;
    const int half = lane >> 4;
    const int n0   = blockIdx.x * 32;
    const int m0   = blockIdx.y * 16;
    const int z    = blockIdx.z;
    const long zq = z / zdiv, zr = z - zq * (long)zdiv;
    A += zq * aHi + zr * aLo;
    B += zq * bHi + zr * bLo;
    C += zq * cHi + zr * cLo;

    const bool transB = (flags & FLAG_TRANSB) != 0;
    const bool addB   = (flags & FLAG_BIAS)   != 0;
    const bool accum  = (flags & FLAG_ACC)    != 0;
    const bool transC = (flags & FLAG_TRANSC) != 0;

    // A fragment row (conv-tap shift, zero-padded via select; address clamped)
    const int  am   = m0 + l15 + rowShiftA;
    const bool aval = (am >= 0) && (am < M);
    const int  amc  = aval ? am : 0;

    // two B fragment columns; second sub-tile may be absent (N % 32 == 16)
    const int  bn0  = n0 + l15;
    const bool n2ok = (n0 + 16) < N;
    const int  bn1  = n2ok ? (bn0 + 16) : bn0;

    const long bcol = transB ? (long)ldb  : (long)ldbe;  // stride per n
    const long bks  = transB ? (long)ldbe : (long)ldb;   // stride per k
    const long bstep = 4 * bks;

    const int kf = 2 * half;                // this lane's K pair in the step
    const float* ap  = A + (long)amc * lda + kf;
    const float* bp0 = B + (long)bn0 * bcol + (long)kf * bks;
    const float* bp1 = B + (long)bn1 * bcol + (long)kf * bks;

    // ---- accumulator init ----
    v8f c0, c1;
    if (accum) {
#pragma unroll
        for (int r = 0; r < 8; ++r) {
            const int cm = m0 + r + 8 * half;
            c0[r] = transC ? C[(long)bn0 * ldc + cm] : C[(long)cm * ldc + bn0];
            c1[r] = n2ok ? (transC ? C[(long)(bn0 + 16) * ldc + cm]
                                   : C[(long)cm * ldc + (bn0 + 16)])
                         : 0.0f;
        }
    } else if (addB) {
        const float bv0 = bias[bn0];
        const float bv1 = n2ok ? bias[bn0 + 16] : 0.0f;
#pragma unroll
        for (int r = 0; r < 8; ++r) { c0[r] = bv0; c1[r] = bv1; }
    } else {
#pragma unroll
        for (int r = 0; r < 8; ++r) { c0[r] = 0.0f; c1[r] = 0.0f; }
    }

    // ---- software-pipelined K loop (unconditional loads + select) ----
    v2f a, b0, b1;
    {
        const float t0 = ap[0], t1 = ap[1];     // always-valid (clamped) addr
        a.x  = aval ? t0 : 0.0f;
        a.y  = aval ? t1 : 0.0f;
        b0.x = bp0[0]; b0.y = bp0[bks];
        b1.x = bp1[0]; b1.y = bp1[bks];
    }

    for (int kk = 4; kk < K; kk += 4) {
        ap += 4; bp0 += bstep; bp1 += bstep;
        v2f an, b0n, b1n;
        const float t0 = ap[0], t1 = ap[1];     // unconditional load
        b0n.x = bp0[0]; b0n.y = bp0[bks];
        b1n.x = bp1[0]; b1n.y = bp1[bks];
        an.x  = aval ? t0 : 0.0f;               // v_cndmask, no branch
        an.y  = aval ? t1 : 0.0f;
        c0 = __builtin_amdgcn_wmma_f32_16x16x4_f32(false, a, false, b0,
                                                   (short)0, c0, false, false);
        c1 = __builtin_amdgcn_wmma_f32_16x16x4_f32(false, a, false, b1,
                                                   (short)0, c1, false, false);
        a = an; b0 = b0n; b1 = b1n;
    }
    c0 = __builtin_amdgcn_wmma_f32_16x16x4_f32(false, a, false, b0,
                                               (short)0, c0, false, false);
    c1 = __builtin_amdgcn_wmma_f32_16x16x4_f32(false, a, false, b1,
                                               (short)0, c1, false, false);

    // ---- store ----
#pragma unroll
    for (int r = 0; r < 8; ++r) {
        const int cm = m0 + r + 8 * half;
        const float v0 = c0[r] * alpha;
        if (transC) C[(long)bn0 * ldc + cm] = v0;
        else        C[(long)cm * ldc + bn0] = v0;
        if (n2ok) {
            const float v1 = c1[r] * alpha;
            if (transC) C[(long)(bn0 + 16) * ldc + cm] = v1;
            else        C[(long)cm * ldc + (bn0 + 16)] = v1;
        }
    }
}

static inline int gx32(int N) { return (N + 31) / 32; }

// ---------------- wave32 block reductions ----------------
__device__ inline float block_sum(float v, float* sm) {
#pragma unroll
    for (int o = 16; o > 0; o >>= 1) v += __shfl_xor(v, o, 32);
    const int wid = threadIdx.x >> 5, lid = threadIdx.x & 31;
    const int nw = (blockDim.x + 31) >> 5;
    __syncthreads();
    if (lid == 0) sm[wid] = v;
    __syncthreads();
    if (wid == 0) {
        float s = (lid < nw) ? sm[lid] : 0.0f;
#pragma unroll
        for (int o = 16; o > 0; o >>= 1) s += __shfl_xor(s, o, 32);
        if (lid == 0) sm[0] = s;
    }
    __syncthreads();
    return sm[0];
}

__device__ inline float block_max(float v, float* sm) {
#pragma unroll
    for (int o = 16; o > 0; o >>= 1) v = fmaxf(v, __shfl_xor(v, o, 32));
    const int wid = threadIdx.x >> 5, lid = threadIdx.x & 31;
    const int nw = (blockDim.x + 31) >> 5;
    __syncthreads();
    if (lid == 0) sm[wid] = v;
    __syncthreads();
    if (wid == 0) {
        float s = (lid < nw) ? sm[lid] : -3.4e38f;
#pragma unroll
        for (int o = 16; o > 0; o >>= 1) s = fmaxf(s, __shfl_xor(s, o, 32));
        if (lid == 0) sm[0] = s;
    }
    __syncthreads();
    return sm[0];
}

// ---------------- pos-emb helper (faithful to source: pos=feature, i=row) --
__device__ inline float pos_emb_val(int row, int h) {
    float den = powf(10000.0f, 2.0f * (float)row / (float)Hc); // may -> inf
    float ang = (float)h / den;                                 // inf -> 0
    return ((row & 1) == 0) ? sinf(ang) : cosf(ang);
}

// x[b,s,h] = emb[tok[b,s],h] + pos(s,h); grid = B*S blocks of H threads
__global__ void embed_pos_kernel(const int* __restrict__ tok,
                                 const float* __restrict__ emb,
                                 float* __restrict__ x)
{
    const long bs = blockIdx.x;
    const int  h  = threadIdx.x;
    const int  s  = (int)(bs % Sc);
    const int  t  = tok[bs];
    x[bs * Hc + h] = emb[(long)t * Hc + h] + pos_emb_val(s, h);
}

// softmax over rows of length `width`; one 256-thread block per row
__global__ void softmax_rows_kernel(float* __restrict__ x, int width)
{
    __shared__ float sm[32];
    const long row = blockIdx.x;
    float* p = x + row * (long)width;
    const int t = threadIdx.x, bd = blockDim.x;
    float m = -3.4e38f;
    for (int i = t; i < width; i += bd) m = fmaxf(m, p[i]);
    m = block_max(m, sm);
    float s = 0.0f;
    for (int i = t; i < width; i += bd) { float e = expf(p[i] - m); p[i] = e; s += e; }
    s = block_sum(s, sm);
    const float inv = 1.0f / s;
    for (int i = t; i < width; i += bd) p[i] *= inv;
}

// out[row,:] = LN(a[row,:] + res[row,:]) * g + bt ; blockDim == width
__global__ void resid_ln_kernel(const float* __restrict__ a,
                                const float* __restrict__ res,
                                const float* __restrict__ g,
                                const float* __restrict__ bt,
                                float* __restrict__ out, int width)
{
    __shared__ float sm[32];
    const long row = blockIdx.x;
    const int  t   = threadIdx.x;
    float v = a[row * (long)width + t];
    if (res) v += res[row * (long)width + t];
    const float mu  = block_sum(v, sm) / (float)width;
    const float d   = v - mu;
    const float var = block_sum(d * d, sm) / (float)width;
    const float r   = rsqrtf(var + 1e-5f);
    out[row * (long)width + t] = d * r * g[t] + bt[t];
}

// dpred[row] = dot(x[row,:H], lw) + lb ; one wave per row
__global__ void dp_linear_kernel(const float* __restrict__ x,
                                 const float* __restrict__ lw,
                                 const float* __restrict__ lb,
                                 float* __restrict__ out)
{
    const long row  = blockIdx.x;
    const int  lane = threadIdx.x;
    float s = 0.0f;
    for (int h = lane; h < Hc; h += 32) s += x[row * Hc + h] * lw[h];
#pragma unroll
    for (int o = 16; o > 0; o >>= 1) s += __shfl_xor(s, o, 32);
    if (lane == 0) out[row] = s + lb[0];
}

// cum[b,s] = inclusive cumsum of durations[b,:]
__global__ void cumsum_kernel(const int* __restrict__ dur, int* __restrict__ cum)
{
    const int b = threadIdx.x;
    if (b >= Bc) return;
    int c = 0;
    for (int s = 0; s < Sc; ++s) { c += dur[b * Sc + s]; cum[b * Sc + s] = c; }
}

// xdec[b,t,h] = xenc[b, searchsorted_right(cum[b],t), h] + pos(t,h)
__global__ void length_reg_kernel(const float* __restrict__ xenc,
                                  const int* __restrict__ cum,
                                  float* __restrict__ xdec)
{
    const long bt = blockIdx.x;              // B*L blocks
    const int  h  = threadIdx.x;             // H threads
    const int  b  = (int)(bt / Lc);
    const int  t  = (int)(bt % Lc);
    const int* c = cum + (long)b * Sc;
    int lo = 0, hi = Sc;
    while (lo < hi) { int mid = (lo + hi) >> 1; if (c[mid] <= t) lo = mid + 1; else hi = mid; }
    int idx = lo < Sc ? lo : Sc - 1;
    xdec[bt * Hc + h] = xenc[((long)b * Sc + idx) * Hc + h] + pos_emb_val(t, h);
}

// =====================================================================
// host-side orchestration
// =====================================================================
struct Layer {
    const float *b_o, *cb1, *cb2, *cw1, *cw2,
                *ln1_b, *ln1_g, *ln2_b, *ln2_g, *w_o, *w_qkv;
};

static Layer load_layer(void* const* d_in, int base) {
    Layer L;
    L.b_o   = (const float*)d_in[base + 0];
    L.cb1   = (const float*)d_in[base + 1];
    L.cb2   = (const float*)d_in[base + 2];
    L.cw1   = (const float*)d_in[base + 3];
    L.cw2   = (const float*)d_in[base + 4];
    L.ln1_b = (const float*)d_in[base + 5];
    L.ln1_g = (const float*)d_in[base + 6];
    L.ln2_b = (const float*)d_in[base + 7];
    L.ln2_g = (const float*)d_in[base + 8];
    L.w_o   = (const float*)d_in[base + 9];
    L.w_qkv = (const float*)d_in[base + 10];
    return L;
}

// one FFT block operating in-place on x = [Bc, Ns, Hc]
static void fft_block(hipStream_t st, float* x, int Ns, const Layer& ly,
                      float* qkv, float* obuf, float* mid, float* tmp, float* sc)
{
    const float inv_sqrt_hc = 0.08838834764831845f; // 1/sqrt(128)

    // qkv[b,s, n*384+d] = sum_h x[b,s,h] * w_qkv[n,h,d]   (batched over b,n)
    wmma_gemm_f32<<<dim3(gx32(384), Ns / 16, Bc * NHc), 32, 0, st>>>(
        x, ly.w_qkv, nullptr, qkv,
        Ns, 384, Hc,
        Hc, 384, 1, 768,
        (long)Ns * Hc, 0L, 0L, (long)Hc * 384, (long)Ns * 768, 384L, NHc,
        0, 1.0f, 0);

    // attention, one (b,head) slice at a time (reuses the L*L score buffer)
    for (int z = 0; z < Bc * NHc; ++z) {
        const int b = z / NHc, n = z % NHc;
        const float* qp = qkv + (long)b * Ns * 768 + n * 384;
        const float* kp = qp + HCc;
        const float* vp = qp + 2 * HCc;
        // scores = (q @ k^T) / sqrt(HC)
        wmma_gemm_f32<<<dim3(gx32(Ns), Ns / 16, 1), 32, 0, st>>>(
            qp, kp, nullptr, sc,
            Ns, Ns, HCc,
            768, 768, 1, Ns,
            0L, 0L, 0L, 0L, 0L, 0L, 1,
            0, inv_sqrt_hc, FLAG_TRANSB);
        softmax_rows_kernel<<<Ns, 256, 0, st>>>(sc, Ns);
        // o[:, n*HC : (n+1)*HC] = attn @ v  (concat heads directly)
        wmma_gemm_f32<<<dim3(gx32(HCc), Ns / 16, 1), 32, 0, st>>>(
            sc, vp, nullptr, obuf + (long)b * Ns * Hc + n * HCc,
            Ns, HCc, Ns,
            Ns, 768, 1, Hc,
            0L, 0L, 0L, 0L, 0L, 0L, 1,
            0, 1.0f, 0);
    }

    // tmp = o @ w_o + b_o ; x = LN(tmp + x)
    wmma_gemm_f32<<<dim3(gx32(Hc), (Bc * Ns) / 16, 1), 32, 0, st>>>(
        obuf, ly.w_o, ly.b_o, tmp,
        Bc * Ns, Hc, Hc,
        Hc, Hc, 1, Hc,
        0L, 0L, 0L, 0L, 0L, 0L, 1,
        0, 1.0f, FLAG_BIAS);
    resid_ln_kernel<<<Bc * Ns, Hc, 0, st>>>(tmp, x, ly.ln1_g, ly.ln1_b, x, Hc);

    // conv1 (H->F, k=3) as 3 shifted GEMMs accumulating into mid
    for (int k = 0; k < KSc; ++k) {
        int fl = FLAG_TRANSB | (k == 0 ? FLAG_BIAS : FLAG_ACC);
        wmma_gemm_f32<<<dim3(gx32(Fc), Ns / 16, Bc), 32, 0, st>>>(
            x, ly.cw1 + k, ly.cb1, mid,
            Ns, Fc, Hc,
            Hc, Hc * KSc, KSc, Fc,
            (long)Ns * Hc, 0L, 0L, 0L, (long)Ns * Fc, 0L, 1,
            k - 1, 1.0f, fl);
    }
    // conv2 (F->H, k=3) into tmp; x = LN(tmp + x)
    for (int k = 0; k < KSc; ++k) {
        int fl = FLAG_TRANSB | (k == 0 ? FLAG_BIAS : FLAG_ACC);
        wmma_gemm_f32<<<dim3(gx32(Hc), Ns / 16, Bc), 32, 0, st>>>(
            mid, ly.cw2 + k, ly.cb2, tmp,
            Ns, Hc, Fc,
            Fc, Fc * KSc, KSc, Hc,
            (long)Ns * Fc, 0L, 0L, 0L, (long)Ns * Hc, 0L, 1,
            k - 1, 1.0f, fl);
    }
    resid_ln_kernel<<<Bc * Ns, Hc, 0, st>>>(tmp, x, ly.ln2_g, ly.ln2_b, x, Hc);
}

extern "C" void kernel_launch(void* const* d_in, const int* in_sizes, int n_in,
                              void* d_out, int out_size, void* d_ws, size_t ws_size,
                              hipStream_t stream)
{
    (void)in_sizes; (void)n_in; (void)out_size; (void)ws_size;

    // JAX pytree flattening: dict keys sorted at every level.
    // 0:tokens 1:durations | 2..45 dec(4 x 11) | 46..55 dp | 56 emb |
    // 57..100 enc(4 x 11) | 101 out_w
    const int* tokens    = (const int*)d_in[0];
    const int* durations = (const int*)d_in[1];
    Layer dec[4], enc[4];
    for (int i = 0; i < 4; ++i) dec[i] = load_layer(d_in, 2 + 11 * i);
    const float* dp_b1  = (const float*)d_in[46];
    const float* dp_b2  = (const float*)d_in[47];
    const float* dp_bt1 = (const float*)d_in[48];
    const float* dp_bt2 = (const float*)d_in[49];
    const float* dp_g1  = (const float*)d_in[50];
    const float* dp_g2  = (const float*)d_in[51];
    const float* dp_lb  = (const float*)d_in[52];
    const float* dp_lw  = (const float*)d_in[53];
    const float* dp_w1  = (const float*)d_in[54];
    const float* dp_w2  = (const float*)d_in[55];
    const float* emb    = (const float*)d_in[56];
    for (int i = 0; i < 4; ++i) enc[i] = load_layer(d_in, 57 + 11 * i);
    const float* out_w  = (const float*)d_in[101];

    // workspace layout (floats)
    float* ws = (float*)d_ws;
    size_t o = 0;
    float* XENC = ws + o; o += (size_t)Bc * Sc * Hc;   // [B,S,H]
    float* XDEC = ws + o; o += (size_t)Bc * Lc * Hc;   // [B,L,H]
    float* QKV  = ws + o; o += (size_t)Bc * Lc * 768;  // [B,Ns,768]
    float* OB   = ws + o; o += (size_t)Bc * Lc * Hc;   // attn out
    float* MID  = ws + o; o += (size_t)Bc * Lc * Fc;   // conv hidden
    float* TMP  = ws + o; o += (size_t)Bc * Lc * Hc;   // pre-LN
    float* SC   = ws + o; o += (size_t)Lc * Lc;        // one (b,h) score slice
    int*   CUM  = (int*)(ws + o);                      // [B,S] cumsum

    float* mel   = (float*)d_out;                      // [B,OUT,L]
    float* dpred = mel + (size_t)Bc * OUTc * Lc;       // [B,S]

    // 1) embedding + positional encoding
    embed_pos_kernel<<<Bc * Sc, Hc, 0, stream>>>(tokens, emb, XENC);

    // 2) encoder
    for (int i = 0; i < 4; ++i)
        fft_block(stream, XENC, Sc, enc[i], QKV, OB, MID, TMP, SC);

    // 3) duration predictor (on encoder output)
    for (int k = 0; k < KSc; ++k) {
        int fl = FLAG_TRANSB | (k == 0 ? FLAG_BIAS : FLAG_ACC);
        wmma_gemm_f32<<<dim3(gx32(Fc), Sc / 16, Bc), 32, 0, stream>>>(
            XENC, dp_w1 + k, dp_b1, MID,
            Sc, Fc, Hc,
            Hc, Hc * KSc, KSc, Fc,
            (long)Sc * Hc, 0L, 0L, 0L, (long)Sc * Fc, 0L, 1,
            k - 1, 1.0f, fl);
    }
    resid_ln_kernel<<<Bc * Sc, Fc, 0, stream>>>(MID, nullptr, dp_g1, dp_bt1, MID, Fc);
    for (int k = 0; k < KSc; ++k) {
        int fl = FLAG_TRANSB | (k == 0 ? FLAG_BIAS : FLAG_ACC);
        wmma_gemm_f32<<<dim3(gx32(Hc), Sc / 16, Bc), 32, 0, stream>>>(
            MID, dp_w2 + k, dp_b2, TMP,
            Sc, Hc, Fc,
            Fc, Fc * KSc, KSc, Hc,
            (long)Sc * Fc, 0L, 0L, 0L, (long)Sc * Hc, 0L, 1,
            k - 1, 1.0f, fl);
    }
    resid_ln_kernel<<<Bc * Sc, Hc, 0, stream>>>(TMP, nullptr, dp_g2, dp_bt2, TMP, Hc);
    dp_linear_kernel<<<Bc * Sc, 32, 0, stream>>>(TMP, dp_lw, dp_lb, dpred);

    // 4) length regulator (+ decoder positional encoding fused)
    cumsum_kernel<<<1, 32, 0, stream>>>(durations, CUM);
    length_reg_kernel<<<Bc * Lc, Hc, 0, stream>>>(XENC, CUM, XDEC);

    // 5) decoder
    for (int i = 0; i < 4; ++i)
        fft_block(stream, XDEC, Lc, dec[i], QKV, OB, MID, TMP, SC);

    // 6) mel projection, stored transposed: mel[b, out, l]
    wmma_gemm_f32<<<dim3(gx32(OUTc), Lc / 16, Bc), 32, 0, stream>>>(
        XDEC, out_w, nullptr, mel,
        Lc, OUTc, Hc,
        Hc, OUTc, 1, Lc,
        (long)Lc * Hc, 0L, 0L, 0L, (long)OUTc * Lc, 0L, 1,
        0, 1.0f, FLAG_TRANSC);
}